// YOLOv8Loss_42829413876164
// MI455X (gfx1250) — compile-verified
//
#include <hip/hip_runtime.h>
#include <hip/hip_bf16.h>
#include <stdint.h>

#define NCLS 6
#define RMAX 16
#define GNT  20
#define BSZ  32
#define NA   8400
#define HW0  6400
#define HW1  1600
#define HW2  400
#define CCH  70
#define NB_LOSS 1050   // BSZ*NA/256

typedef __attribute__((ext_vector_type(2))) float v2f;
typedef __attribute__((ext_vector_type(8))) float v8f;

// ---------------- device helpers ----------------

__device__ __forceinline__ void anchor_decode(int a, int& scale, int& off, int& HW,
                                              float& stride, float& ax, float& ay) {
  int W;
  if (a < HW0)             { scale = 0; off = a;            W = 80; HW = HW0; stride = 8.f;  }
  else if (a < HW0 + HW1)  { scale = 1; off = a - HW0;      W = 40; HW = HW1; stride = 16.f; }
  else                     { scale = 2; off = a - HW0 - HW1; W = 20; HW = HW2; stride = 32.f; }
  int y = off / W, x = off - y * W;
  ax = ((float)x + 0.5f) * stride;
  ay = ((float)y + 0.5f) * stride;
}

__device__ __forceinline__ float iou_plain(float ax1, float ay1, float ax2, float ay2,
                                           float bx1, float by1, float bx2, float by2) {
  const float eps = 1e-7f;
  float ix = fmaxf(fminf(ax2, bx2) - fmaxf(ax1, bx1), 0.f);
  float iy = fmaxf(fminf(ay2, by2) - fmaxf(ay1, by1), 0.f);
  float inter = ix * iy;
  float w1 = fmaxf(ax2 - ax1, eps), h1 = fmaxf(ay2 - ay1, eps);
  float w2 = fmaxf(bx2 - bx1, eps), h2 = fmaxf(by2 - by1, eps);
  float uni = w1 * h1 + w2 * h2 - inter + eps;
  return inter / uni;
}

// b1 = prediction, b2 = target (matches reference _iou(pd, tgt, ciou=True))
__device__ __forceinline__ float iou_ciou(float ax1, float ay1, float ax2, float ay2,
                                          float bx1, float by1, float bx2, float by2) {
  const float eps = 1e-7f;
  float iou = iou_plain(ax1, ay1, ax2, ay2, bx1, by1, bx2, by2);
  float w1 = fmaxf(ax2 - ax1, eps), h1 = fmaxf(ay2 - ay1, eps);
  float w2 = fmaxf(bx2 - bx1, eps), h2 = fmaxf(by2 - by1, eps);
  float cw = fmaxf(ax2, bx2) - fminf(ax1, bx1);
  float ch = fmaxf(ay2, by2) - fminf(ay1, by1);
  float c2 = cw * cw + ch * ch + eps;
  float dx = ax1 + ax2 - bx1 - bx2;
  float dy = ay1 + ay2 - by1 - by2;
  float rho2 = (dx * dx + dy * dy) * 0.25f;
  const float k = 4.f / (3.14159265358979323846f * 3.14159265358979323846f);
  float da = atanf(w2 / h2) - atanf(w1 / h1);
  float v = k * da * da;
  float alpha = v / (v - iou + (1.f + eps));
  return iou - (rho2 / c2 + v * alpha);
}

// ---------------- kernels ----------------

// K0: parse targets -> gt boxes (xyxy, pixels) + labels
__global__ void prep_kernel(const float* __restrict__ targets,
                            float* __restrict__ gtbox, int* __restrict__ gtlab) {
  int r = blockIdx.x * blockDim.x + threadIdx.x;
  if (r >= BSZ * GNT) return;
  const float* t = targets + (size_t)r * 6;
  gtlab[r] = (int)t[1];
  float cx = t[2] * 640.f, cy = t[3] * 640.f, w = t[4] * 640.f, h = t[5] * 640.f;
  gtbox[r * 4 + 0] = cx - 0.5f * w;
  gtbox[r * 4 + 1] = cy - 0.5f * h;
  gtbox[r * 4 + 2] = cx + 0.5f * w;
  gtbox[r * 4 + 3] = cy + 0.5f * h;
}

// K1: DFL decode -> pred boxes [BSZ*NA*4]
__global__ void decode_kernel(const float* __restrict__ p0, const float* __restrict__ p1,
                              const float* __restrict__ p2, float* __restrict__ pdbox) {
  int idx = blockIdx.x * blockDim.x + threadIdx.x;   // exact fit: 1050*256 = BSZ*NA
  int b = idx / NA, a = idx - b * NA;
  int scale, off, HW; float stride, ax, ay;
  anchor_decode(a, scale, off, HW, stride, ax, ay);
  const float* P = (scale == 0) ? p0 : ((scale == 1) ? p1 : p2);
  size_t cbase = (size_t)b * CCH * HW + off;
  float d[4];
#pragma unroll
  for (int s = 0; s < 4; ++s) {
    float vals[RMAX], m = -1e30f;
#pragma unroll
    for (int k = 0; k < RMAX; ++k) {
      float x = P[cbase + (size_t)(s * RMAX + k) * HW];
      vals[k] = x; m = fmaxf(m, x);
    }
    float se = 0.f, sw = 0.f;
#pragma unroll
    for (int k = 0; k < RMAX; ++k) {
      float e = expf(vals[k] - m);
      se += e; sw += e * (float)k;
    }
    d[s] = sw / se;
  }
  pdbox[(size_t)idx * 4 + 0] = ax - d[0] * stride;
  pdbox[(size_t)idx * 4 + 1] = ay - d[1] * stride;
  pdbox[(size_t)idx * 4 + 2] = ax + d[2] * stride;
  pdbox[(size_t)idx * 4 + 3] = ay + d[3] * stride;
}

// K2: align metric [BSZ*GNT, NA]; one block per (b,g); GT box staged via async-to-LDS
__global__ void align_kernel(const float* __restrict__ p0, const float* __restrict__ p1,
                             const float* __restrict__ p2, const float* __restrict__ pdbox,
                             const float* __restrict__ gtbox, const int* __restrict__ gtlab,
                             float* __restrict__ alignm) {
  int r = blockIdx.x;          // b*GNT + g
  int b = r / GNT;
  __shared__ __align__(16) float sbox[4];
  __shared__ int slab;
  if (threadIdx.x == 0) {
    unsigned ldsa = (unsigned)(size_t)(&sbox[0]);
    const float* ga = gtbox + (size_t)r * 4;
    // CDNA5 async memory->LDS copy (ASYNCcnt-tracked)
    asm volatile("global_load_async_to_lds_b128 %0, %1, off"
                 :: "v"(ldsa), "v"(ga) : "memory");
    slab = gtlab[r];
  }
  asm volatile("s_wait_asynccnt 0" ::: "memory");
  __syncthreads();
  float gx1 = sbox[0], gy1 = sbox[1], gx2 = sbox[2], gy2 = sbox[3];
  int lab = slab;
  for (int a = threadIdx.x; a < NA; a += blockDim.x) {
    int scale, off, HW; float stride, ax, ay;
    anchor_decode(a, scale, off, HW, stride, ax, ay);
    const float* P = (scale == 0) ? p0 : ((scale == 1) ? p1 : p2);
    float logit = P[((size_t)b * CCH + 64 + lab) * HW + off];
    float sc = 1.f / (1.f + expf(-logit));
    const float* pb = pdbox + ((size_t)b * NA + a) * 4;
    __builtin_prefetch(pb + 4 * 256, 0, 1);   // global_prefetch_b8, streaming hint
    float iou = iou_plain(gx1, gy1, gx2, gy2, pb[0], pb[1], pb[2], pb[3]);
    float ing = fminf(fminf(ax - gx1, ay - gy1), fminf(gx2 - ax, gy2 - ay));
    float al = 0.f;
    if (ing > 1e-9f) {
      float i2 = iou * iou;
      al = sqrtf(sc) * (i2 * i2 * i2);
    }
    alignm[(size_t)r * NA + a] = al;
  }
}

// merge two sorted-descending top-10 lists, keep top-10 (dst may alias a)
__device__ __forceinline__ void merge10(float* dst, const float* a, const float* b) {
  float out[10];
  int i = 0, j = 0;
#pragma unroll
  for (int k = 0; k < 10; ++k) {
    bool ta = (j >= 10) || (i < 10 && a[i] >= b[j]);
    out[k] = ta ? a[i++] : b[j++];
  }
#pragma unroll
  for (int k = 0; k < 10; ++k) dst[k] = out[k];
}

// K3: per (b,g) row: top-10 threshold + row max
__global__ void topk_kernel(const float* __restrict__ alignm,
                            float* __restrict__ thresh, float* __restrict__ amax) {
  int r = blockIdx.x;
  const float* row = alignm + (size_t)r * NA;
  float t[10];
#pragma unroll
  for (int k = 0; k < 10; ++k) t[k] = -1e30f;
  for (int a = threadIdx.x; a < NA; a += blockDim.x) {
    float v = row[a];
    if (v > t[9]) {
      int i = 9;
      while (i > 0 && v > t[i - 1]) { t[i] = t[i - 1]; --i; }
      t[i] = v;
    }
  }
  __shared__ float buf[256 * 10];
#pragma unroll
  for (int k = 0; k < 10; ++k) buf[threadIdx.x * 10 + k] = t[k];
  for (int s = 128; s > 0; s >>= 1) {
    __syncthreads();
    if ((int)threadIdx.x < s)
      merge10(buf + threadIdx.x * 10, buf + threadIdx.x * 10, buf + (threadIdx.x + s) * 10);
  }
  __syncthreads();
  if (threadIdx.x == 0) { amax[r] = buf[0]; thresh[r] = buf[9]; }
}

// K4: per-anchor GT bitmask with multi-assignment resolution (argmax of align)
__global__ void mask_kernel(const float* __restrict__ alignm, const float* __restrict__ thresh,
                            unsigned* __restrict__ maskb) {
  int idx = blockIdx.x * blockDim.x + threadIdx.x;   // b*NA + a
  int b = idx / NA, a = idx - b * NA;
  unsigned bits = 0; int cnt = 0;
  float best = -1.f; int bestg = 0;
#pragma unroll
  for (int g = 0; g < GNT; ++g) {
    int r = b * GNT + g;
    float v = alignm[(size_t)r * NA + a];
    if (v >= thresh[r] && v > 0.f) { bits |= (1u << g); ++cnt; }
    if (v > best) { best = v; bestg = g; }   // first max on ties
  }
  if (cnt > 1) bits = (1u << bestg);
  maskb[idx] = bits;
}

// K5: fused losses. All four wave-level reductions are done by TWO
// V_WMMA_F32_16X16X4_F32 ops (matrix pipe) instead of ds_bpermute ladders:
//   A slot layout (16x4 f32): lane l<16 -> (M=l,K=0),(M=l,K=1);
//                             lane l>=16 -> (M=l-16,K=2),(M=l-16,K=3)
//   B selector (4x16): B[0][0]=B[2][0]=1, B[1][1]=B[3][1]=1, else 0
//   => D[m,0] = X_m + X_{m+16},  D[m,1] = Y_m + Y_{m+16}
//   D column 0 lives in lanes {0,16} (row halves), column 1 in lanes {1,17}.
__global__ void loss_kernel(const float* __restrict__ p0, const float* __restrict__ p1,
                            const float* __restrict__ p2, const float* __restrict__ pdbox,
                            const float* __restrict__ gtbox, const int* __restrict__ gtlab,
                            const float* __restrict__ alignm, const float* __restrict__ amax,
                            const unsigned* __restrict__ maskb, float* __restrict__ partials) {
  int idx = blockIdx.x * blockDim.x + threadIdx.x;   // exact fit, no guard (EXEC all-1 for WMMA)
  int b = idx / NA, a = idx - b * NA;
  int scale, off, HW; float stride, ax, ay;
  anchor_decode(a, scale, off, HW, stride, ax, ay);
  const float* P = (scale == 0) ? p0 : ((scale == 1) ? p1 : p2);
  size_t cbase = (size_t)b * CCH * HW + off;

  float px1 = pdbox[(size_t)idx * 4 + 0], py1 = pdbox[(size_t)idx * 4 + 1];
  float px2 = pdbox[(size_t)idx * 4 + 2], py2 = pdbox[(size_t)idx * 4 + 3];

  unsigned bits = maskb[idx];
  bool fg = bits != 0u;
  int g = fg ? (__ffs(bits) - 1) : 0;
  float sval = 0.f; int lab = 0;
  float tx1 = 0.f, ty1 = 0.f, tx2 = 0.f, ty2 = 0.f;
  if (fg) {
    int r = b * GNT + g;
    tx1 = gtbox[r * 4 + 0]; ty1 = gtbox[r * 4 + 1];
    tx2 = gtbox[r * 4 + 2]; ty2 = gtbox[r * 4 + 3];
    lab = gtlab[r];
    float i0 = fmaxf(iou_plain(px1, py1, px2, py2, tx1, ty1, tx2, ty2), 0.f);
    float ng = alignm[(size_t)r * NA + a] / (amax[r] + 1e-9f);
    sval = ng * i0;
  }

  // BCE-with-logits over all classes (dominant term, all anchors)
  float cls_local = 0.f;
#pragma unroll
  for (int c = 0; c < NCLS; ++c) {
    float x = P[cbase + (size_t)(64 + c) * HW];
    float ts = (fg && c == lab) ? sval : 0.f;
    cls_local += fmaxf(x, 0.f) - x * ts + log1pf(expf(-fabsf(x)));
  }

  float bbox_local = 0.f, dfl_local = 0.f, np_local = 0.f;
  if (fg) {
    np_local = 1.f;
    bbox_local = 1.f - iou_ciou(px1, py1, px2, py2, tx1, ty1, tx2, ty2);
    float ltrb[4] = { (ax - tx1) / stride, (ay - ty1) / stride,
                      (tx2 - ax) / stride, (ty2 - ay) / stride };
#pragma unroll
    for (int s = 0; s < 4; ++s) {
      float v = fminf(fmaxf(ltrb[s], 0.f), (float)RMAX - 1.01f);
      int tl = (int)v;
      float wl = (float)(tl + 1) - v;
      float wr = v - (float)tl;
      float vals[RMAX], m = -1e30f;
#pragma unroll
      for (int k = 0; k < RMAX; ++k) {
        float x = P[cbase + (size_t)(s * RMAX + k) * HW];
        vals[k] = x; m = fmaxf(m, x);
      }
      float se = 0.f;
#pragma unroll
      for (int k = 0; k < RMAX; ++k) se += expf(vals[k] - m);
      float logZ = m + logf(se);
      dfl_local += (logZ - vals[tl]) * wl + (logZ - vals[tl + 1]) * wr;
    }
  }

  // ---- WMMA wave reductions (all lanes converged, EXEC all-1) ----
  int lane = threadIdx.x & 31;
  int n = lane & 15;
  v2f Bm; Bm.x = (n == 0) ? 1.f : 0.f;
          Bm.y = (n == 1) ? 1.f : 0.f;
  v2f A1; A1.x = cls_local;  A1.y = bbox_local;
  v2f A2; A2.x = dfl_local;  A2.y = np_local;
  v8f Z = {0.f, 0.f, 0.f, 0.f, 0.f, 0.f, 0.f, 0.f};
  v8f C1 = __builtin_amdgcn_wmma_f32_16x16x4_f32(false, A1, false, Bm, (short)0, Z, false, false);
  v8f C2 = __builtin_amdgcn_wmma_f32_16x16x4_f32(false, A2, false, Bm, (short)0, Z, false, false);
  float ps1 = C1[0] + C1[1] + C1[2] + C1[3] + C1[4] + C1[5] + C1[6] + C1[7];
  float ps2 = C2[0] + C2[1] + C2[2] + C2[3] + C2[4] + C2[5] + C2[6] + C2[7];

  __shared__ float sw[8][2][4];
  int wave = threadIdx.x >> 5, half = lane >> 4;
  if (n == 0) { sw[wave][half][0] = ps1; sw[wave][half][2] = ps2; }  // cls, dfl
  if (n == 1) { sw[wave][half][1] = ps1; sw[wave][half][3] = ps2; }  // bbox, np
  __syncthreads();
  if (threadIdx.x == 0) {
    float s0 = 0.f, s1 = 0.f, s2 = 0.f, s3 = 0.f;
#pragma unroll
    for (int w = 0; w < 8; ++w)
#pragma unroll
      for (int h = 0; h < 2; ++h) {
        s0 += sw[w][h][0]; s1 += sw[w][h][1]; s2 += sw[w][h][2]; s3 += sw[w][h][3];
      }
    partials[(size_t)blockIdx.x * 4 + 0] = s0;
    partials[(size_t)blockIdx.x * 4 + 1] = s1;
    partials[(size_t)blockIdx.x * 4 + 2] = s2;
    partials[(size_t)blockIdx.x * 4 + 3] = s3;
  }
}

// K6: ordered (deterministic) final reduction + weighting
__global__ void finalize_kernel(const float* __restrict__ partials, float* __restrict__ out) {
  if (blockIdx.x != 0 || threadIdx.x != 0) return;
  float cls = 0.f, bbox = 0.f, dfl = 0.f, np = 0.f;
  for (int i = 0; i < NB_LOSS; ++i) {
    cls  += partials[(size_t)i * 4 + 0];
    bbox += partials[(size_t)i * 4 + 1];
    dfl  += partials[(size_t)i * 4 + 2];
    np   += partials[(size_t)i * 4 + 3];
  }
  float lcls = cls / (float)NA;
  float npos = fmaxf(np, 1.f);
  float lbox = 7.5f * bbox / npos;
  float ldfl = 1.5f * dfl / npos;
  out[0] = lcls + lbox + ldfl;
  out[1] = lcls;
  out[2] = lbox;
  out[3] = 0.f;
  out[4] = ldfl;
}

// ---------------- launch ----------------

extern "C" void kernel_launch(void* const* d_in, const int* in_sizes, int n_in,
                              void* d_out, int out_size, void* d_ws, size_t ws_size,
                              hipStream_t stream) {
  (void)in_sizes; (void)n_in; (void)out_size; (void)ws_size;
  const float* p0 = (const float*)d_in[0];
  const float* p1 = (const float*)d_in[1];
  const float* p2 = (const float*)d_in[2];
  const float* targets = (const float*)d_in[3];

  char* ws = (char*)d_ws;
  size_t o = 0;
  auto take = [&](size_t nbytes) -> void* {
    void* p = (void*)(ws + o);
    o += (nbytes + 255) & ~(size_t)255;
    return p;
  };
  float*    pdbox   = (float*)take((size_t)BSZ * NA * 4 * sizeof(float));
  float*    gtbox   = (float*)take((size_t)BSZ * GNT * 4 * sizeof(float));
  int*      gtlab   = (int*)take((size_t)BSZ * GNT * sizeof(int));
  float*    alignm  = (float*)take((size_t)BSZ * GNT * NA * sizeof(float));
  float*    thresh  = (float*)take((size_t)BSZ * GNT * sizeof(float));
  float*    amax    = (float*)take((size_t)BSZ * GNT * sizeof(float));
  unsigned* maskb   = (unsigned*)take((size_t)BSZ * NA * sizeof(unsigned));
  float*    partial = (float*)take((size_t)NB_LOSS * 4 * sizeof(float));

  prep_kernel   <<<5, 128, 0, stream>>>(targets, gtbox, gtlab);
  decode_kernel <<<NB_LOSS, 256, 0, stream>>>(p0, p1, p2, pdbox);
  align_kernel  <<<BSZ * GNT, 256, 0, stream>>>(p0, p1, p2, pdbox, gtbox, gtlab, alignm);
  topk_kernel   <<<BSZ * GNT, 256, 0, stream>>>(alignm, thresh, amax);
  mask_kernel   <<<NB_LOSS, 256, 0, stream>>>(alignm, thresh, maskb);
  loss_kernel   <<<NB_LOSS, 256, 0, stream>>>(p0, p1, p2, pdbox, gtbox, gtlab,
                                              alignm, amax, maskb, partial);
  finalize_kernel<<<1, 32, 0, stream>>>(partial, (float*)d_out);
}